// GeometricDihedralMessagePassing_45973329936787
// MI455X (gfx1250) — compile-verified
//
#include <hip/hip_runtime.h>
#include <math.h>

#define N_NODES   100000
#define SCALAR_DIM 64
#define HIDDEN     64
#define ATTR_DIM    3
#define N_DIH   1000000
#define IN_DIM    133           // 64*2 + 3 + 2
#define K1PAD     136           // IN_DIM padded to multiple of 4 (zeros)
#define INP_LD    140           // s_inp row pitch: banks (12m+k)%64 all distinct
#define H_LD       68           // hidden row pitch: banks (4m+k)%64 all distinct
#define WP         80           // weight pair pitch: h=0/h=1 use disjoint bank halves
#define EPS_LN    1e-5f
#define EPS_NORM  1e-6f
#define WAVES       4           // waves per block (wave32)
#define TPW         4           // tiles (of 16 dihedrals) per wave
#define BLOCK     (WAVES * 32)

typedef __attribute__((ext_vector_type(2))) float v2f;
typedef __attribute__((ext_vector_type(8))) float v8f;

__device__ __forceinline__ v8f wmma_f32_16x16x4(v2f a, v2f b, v8f c) {
  // D = A(16x4,f32) * B(4x16,f32) + C(16x16,f32), full fp32 precision
  return __builtin_amdgcn_wmma_f32_16x16x4_f32(false, a, false, b, (short)0, c,
                                               false, false);
}

// ---------------------------------------------------------------------------
// Kernel 1: zero the workspace (segment sums + counts)
// ---------------------------------------------------------------------------
__global__ void zero_ws_kernel(float* __restrict__ p, long n) {
  long i = (long)blockIdx.x * blockDim.x + threadIdx.x;
  long stride = (long)gridDim.x * blockDim.x;
  for (; i < n; i += stride) p[i] = 0.0f;
}

// ---------------------------------------------------------------------------
// Kernel 2: per-dihedral geometry + LayerNorm + 3-layer MLP (WMMA fp32)
//           + atomic scatter of messages into per-node segment sums.
// One wave handles TPW tiles of 16 dihedrals; weights staged once per block
// in a k-pair-swizzled LDS layout so every B fragment is one ds_load_b64.
// ---------------------------------------------------------------------------
__global__ __launch_bounds__(BLOCK)
void dih_msg_kernel(const float* __restrict__ x,
                    const float* __restrict__ pos,
                    const int*   __restrict__ qidx,   // (4, N_DIH)
                    const float* __restrict__ qattr,  // (N_DIH, 3)
                    const float* __restrict__ gamma,
                    const float* __restrict__ beta,
                    const float* __restrict__ w1, const float* __restrict__ b1,
                    const float* __restrict__ w2, const float* __restrict__ b2,
                    const float* __restrict__ w3, const float* __restrict__ b3,
                    float* __restrict__ sum_j, float* __restrict__ sum_k,
                    float* __restrict__ cnt_j, float* __restrict__ cnt_k)
{
  // block-shared weights, swizzled: s[(kq*WP + n)*2 + e] = w[(2kq+e)][n]
  __shared__ __align__(16) float s_w1[(K1PAD / 2) * WP * 2];   // 43.5 KB
  __shared__ __align__(16) float s_w2[(HIDDEN / 2) * WP * 2];  // 20.5 KB
  __shared__ __align__(16) float s_w3[(HIDDEN / 2) * WP * 2];  // 20.5 KB
  __shared__ float s_gamma[K1PAD], s_beta[K1PAD];
  // per-wave staging (bank-conflict-free pitches)
  __shared__ __align__(16) float s_inp[WAVES][16][INP_LD];
  __shared__ __align__(16) float s_hA [WAVES][16][H_LD];
  __shared__ __align__(16) float s_hB [WAVES][16][H_LD];
  __shared__ int s_i[WAVES][16], s_l[WAVES][16];
  __shared__ int s_j[WAVES][16], s_k[WAVES][16], s_v[WAVES][16];

  const int tid  = threadIdx.x;
  const int lane = tid & 31;
  const int wv   = tid >> 5;

  // ---- Stage W: cooperative swizzled weight staging (once per block) ------
  for (int p = tid; p < (K1PAD / 2) * HIDDEN; p += BLOCK) {
    int kq = p >> 6, n = p & 63;
    int k0 = kq * 2;
    v2f e;
    e.x = (k0     < IN_DIM) ? w1[k0       * HIDDEN + n] : 0.0f;
    e.y = (k0 + 1 < IN_DIM) ? w1[(k0 + 1) * HIDDEN + n] : 0.0f;
    *(v2f*)&s_w1[(kq * WP + n) * 2] = e;
  }
  for (int p = tid; p < (HIDDEN / 2) * HIDDEN; p += BLOCK) {
    int kq = p >> 6, n = p & 63;
    int k0 = kq * 2;
    v2f e2; e2.x = w2[k0 * HIDDEN + n]; e2.y = w2[(k0 + 1) * HIDDEN + n];
    *(v2f*)&s_w2[(kq * WP + n) * 2] = e2;
    v2f e3; e3.x = w3[k0 * HIDDEN + n]; e3.y = w3[(k0 + 1) * HIDDEN + n];
    *(v2f*)&s_w3[(kq * WP + n) * 2] = e3;
  }
  for (int c = tid; c < K1PAD; c += BLOCK) {
    s_gamma[c] = (c < IN_DIM) ? gamma[c] : 0.0f;
    s_beta[c]  = (c < IN_DIM) ? beta[c]  : 0.0f;
  }

  const int mrow  = lane & 15;        // A-fragment row / D column index
  const int hh    = lane >> 4;        // half-wave id
  const int khalf = hh * 2;           // lanes 16..31 carry K+2,K+3

  const long tile0 = ((long)blockIdx.x * WAVES + wv) * TPW;

  for (int it = 0; it < TPW; ++it) {
    const long base = (tile0 + it) * 16;

    // ---- Stage A: geometry, one dihedral per lane (lanes 0..15) -----------
    if (lane < 16) {
      long d = base + lane;
      int valid = (d < N_DIH) ? 1 : 0;
      int ii = 0, jj = 0, kk = 0, ll = 0;
      float cosd = 0.f, sind = 0.f, a0 = 0.f, a1 = 0.f, a2 = 0.f;
      if (valid) {
        ii = qidx[0L * N_DIH + d];
        jj = qidx[1L * N_DIH + d];
        kk = qidx[2L * N_DIH + d];
        ll = qidx[3L * N_DIH + d];
        float pix = pos[(long)ii*3+0], piy = pos[(long)ii*3+1], piz = pos[(long)ii*3+2];
        float pjx = pos[(long)jj*3+0], pjy = pos[(long)jj*3+1], pjz = pos[(long)jj*3+2];
        float pkx = pos[(long)kk*3+0], pky = pos[(long)kk*3+1], pkz = pos[(long)kk*3+2];
        float plx = pos[(long)ll*3+0], ply = pos[(long)ll*3+1], plz = pos[(long)ll*3+2];
        float d1x = pjx-pix, d1y = pjy-piy, d1z = pjz-piz;
        float d2x = pkx-pjx, d2y = pky-pjy, d2z = pkz-pjz;
        float d3x = plx-pkx, d3y = ply-pky, d3z = plz-pkz;
        float n1x = d1y*d2z - d1z*d2y, n1y = d1z*d2x - d1x*d2z, n1z = d1x*d2y - d1y*d2x;
        float n2x = d2y*d3z - d2z*d3y, n2y = d2z*d3x - d2x*d3z, n2z = d2x*d3y - d2y*d3x;
        float in1 = 1.0f / fmaxf(sqrtf(n1x*n1x + n1y*n1y + n1z*n1z), EPS_NORM);
        float in2 = 1.0f / fmaxf(sqrtf(n2x*n2x + n2y*n2y + n2z*n2z), EPS_NORM);
        n1x *= in1; n1y *= in1; n1z *= in1;
        n2x *= in2; n2y *= in2; n2z *= in2;
        cosd = fminf(fmaxf(n1x*n2x + n1y*n2y + n1z*n2z, -1.0f), 1.0f);
        float ib2 = 1.0f / fmaxf(sqrtf(d2x*d2x + d2y*d2y + d2z*d2z), EPS_NORM);
        float ux = d2x*ib2, uy = d2y*ib2, uz = d2z*ib2;
        float mx = n1y*uz - n1z*uy, my = n1z*ux - n1x*uz, mz = n1x*uy - n1y*ux;
        sind = mx*n2x + my*n2y + mz*n2z;
        a0 = qattr[d*3+0]; a1 = qattr[d*3+1]; a2 = qattr[d*3+2];
        atomicAdd(&cnt_j[jj], 1.0f);
        atomicAdd(&cnt_k[kk], 1.0f);
      }
      s_i[wv][lane] = ii; s_l[wv][lane] = ll;
      s_j[wv][lane] = jj; s_k[wv][lane] = kk; s_v[wv][lane] = valid;
      s_inp[wv][lane][128] = a0;
      s_inp[wv][lane][129] = a1;
      s_inp[wv][lane][130] = a2;
      s_inp[wv][lane][131] = cosd;
      s_inp[wv][lane][132] = sind;
      s_inp[wv][lane][133] = 0.0f;   // zero pad: K-loop needs no tail guard
      s_inp[wv][lane][134] = 0.0f;
      s_inp[wv][lane][135] = 0.0f;
    }
    __syncthreads();

    // ---- Stage B: gather x[i]/x[l] rows, float4 per lane ------------------
    {
      const int c4 = lane & 15;          // float4 chunk within 64-float row
      for (int m = 0; m < 16; ++m) {
        long node = hh ? (long)s_l[wv][m] : (long)s_i[wv][m];
        float4 val = ((const float4*)(x + node * SCALAR_DIM))[c4];
        *(float4*)&s_inp[wv][m][hh * SCALAR_DIM + c4 * 4] = val;
      }
    }
    __syncthreads();

    // ---- Stage C: LayerNorm, full-wave: lanes L / L+16 split row L --------
    {
      float s = 0.0f;
      for (int c = hh; c < IN_DIM; c += 2) s += s_inp[wv][mrow][c];
      s += __shfl_xor(s, 16);
      float mu = s * (1.0f / IN_DIM);
      float v = 0.0f;
      for (int c = hh; c < IN_DIM; c += 2) {
        float t = s_inp[wv][mrow][c] - mu; v += t * t;
      }
      v += __shfl_xor(v, 16);
      float rs = rsqrtf(v * (1.0f / IN_DIM) + EPS_LN);
      for (int c = hh; c < IN_DIM; c += 2)
        s_inp[wv][mrow][c] = (s_inp[wv][mrow][c] - mu) * rs * s_gamma[c] + s_beta[c];
    }
    __syncthreads();

    // ---- GEMM1: [16 x 136] @ [136 x 64], bias + SiLU -> s_hA --------------
    for (int t = 0; t < 4; ++t) {
      v8f acc = {};
      const int n = t * 16 + mrow;
#pragma unroll
      for (int kb = 0; kb < K1PAD; kb += 4) {
        const int k0 = kb + khalf;
        const int kq = (kb >> 1) + hh;
        v2f a = *(const v2f*)&s_inp[wv][mrow][k0];
        v2f b = *(const v2f*)&s_w1[(kq * WP + n) * 2];
        acc = wmma_f32_16x16x4(a, b, acc);
      }
      float bias = b1[n];
#pragma unroll
      for (int r = 0; r < 8; ++r) {
        float v = acc[r] + bias;
        v = v / (1.0f + __expf(-v));               // SiLU
        s_hA[wv][r + hh * 8][n] = v;
      }
    }
    __syncthreads();

    // ---- GEMM2: [16 x 64] @ [64 x 64], bias + SiLU -> s_hB ----------------
    for (int t = 0; t < 4; ++t) {
      v8f acc = {};
      const int n = t * 16 + mrow;
#pragma unroll
      for (int kb = 0; kb < HIDDEN; kb += 4) {
        const int k0 = kb + khalf;
        const int kq = (kb >> 1) + hh;
        v2f a = *(const v2f*)&s_hA[wv][mrow][k0];
        v2f b = *(const v2f*)&s_w2[(kq * WP + n) * 2];
        acc = wmma_f32_16x16x4(a, b, acc);
      }
      float bias = b2[n];
#pragma unroll
      for (int r = 0; r < 8; ++r) {
        float v = acc[r] + bias;
        v = v / (1.0f + __expf(-v));               // SiLU
        s_hB[wv][r + hh * 8][n] = v;
      }
    }
    __syncthreads();

    // ---- GEMM3: [16 x 64] @ [64 x 64] + bias, atomic scatter --------------
    for (int t = 0; t < 4; ++t) {
      v8f acc = {};
      const int n = t * 16 + mrow;
#pragma unroll
      for (int kb = 0; kb < HIDDEN; kb += 4) {
        const int k0 = kb + khalf;
        const int kq = (kb >> 1) + hh;
        v2f a = *(const v2f*)&s_hB[wv][mrow][k0];
        v2f b = *(const v2f*)&s_w3[(kq * WP + n) * 2];
        acc = wmma_f32_16x16x4(a, b, acc);
      }
      float bias = b3[n];
#pragma unroll
      for (int r = 0; r < 8; ++r) {
        int m = r + hh * 8;
        if (s_v[wv][m]) {
          float v = acc[r] + bias;
          atomicAdd(&sum_j[(long)s_j[wv][m] * SCALAR_DIM + n], v);
          atomicAdd(&sum_k[(long)s_k[wv][m] * SCALAR_DIM + n], v);
        }
      }
    }
    __syncthreads();
  }
}

// ---------------------------------------------------------------------------
// Kernel 3: per-node aggregation + output GEMM (aggr @ w_out) * (1/sqrt(64))
// One wave handles a tile of 16 nodes.
// ---------------------------------------------------------------------------
__global__ __launch_bounds__(BLOCK)
void node_out_kernel(const float* __restrict__ sum_j,
                     const float* __restrict__ sum_k,
                     const float* __restrict__ cnt_j,
                     const float* __restrict__ cnt_k,
                     const float* __restrict__ w_out,
                     float* __restrict__ out)
{
  __shared__ __align__(16) float s_w[(SCALAR_DIM / 2) * WP * 2];
  __shared__ __align__(16) float s_a[WAVES][16][H_LD];

  const int tid  = threadIdx.x;
  const int lane = tid & 31;
  const int wv   = tid >> 5;
  const long tile = (long)blockIdx.x * WAVES + wv;
  const long base = tile * 16;

  for (int p = tid; p < (SCALAR_DIM / 2) * SCALAR_DIM; p += BLOCK) {
    int kq = p >> 6, n = p & 63;
    int k0 = kq * 2;
    v2f e; e.x = w_out[k0 * SCALAR_DIM + n]; e.y = w_out[(k0 + 1) * SCALAR_DIM + n];
    *(v2f*)&s_w[(kq * WP + n) * 2] = e;
  }

  for (int m = 0; m < 16; ++m) {
    long node = base + m;
    if (node < N_NODES) {
      float rj = 1.0f / fmaxf(cnt_j[node], 1.0f);
      float rk = 1.0f / fmaxf(cnt_k[node], 1.0f);
      for (int c = lane; c < SCALAR_DIM; c += 32)
        s_a[wv][m][c] = 0.5f * (sum_j[node * SCALAR_DIM + c] * rj +
                                sum_k[node * SCALAR_DIM + c] * rk);
    } else {
      for (int c = lane; c < SCALAR_DIM; c += 32) s_a[wv][m][c] = 0.0f;
    }
  }
  __syncthreads();

  const int mrow  = lane & 15;
  const int hh    = lane >> 4;
  const int khalf = hh * 2;
  const float scale = 0.125f;  // 1/sqrt(64)

  for (int t = 0; t < 4; ++t) {
    v8f acc = {};
    const int n = t * 16 + mrow;
#pragma unroll
    for (int kb = 0; kb < SCALAR_DIM; kb += 4) {
      const int k0 = kb + khalf;
      const int kq = (kb >> 1) + hh;
      v2f a = *(const v2f*)&s_a[wv][mrow][k0];
      v2f b = *(const v2f*)&s_w[(kq * WP + n) * 2];
      acc = wmma_f32_16x16x4(a, b, acc);
    }
#pragma unroll
    for (int r = 0; r < 8; ++r) {
      long node = base + r + hh * 8;
      if (node < N_NODES) out[node * SCALAR_DIM + n] = acc[r] * scale;
    }
  }
}

// ---------------------------------------------------------------------------
extern "C" void kernel_launch(void* const* d_in, const int* in_sizes, int n_in,
                              void* d_out, int out_size, void* d_ws, size_t ws_size,
                              hipStream_t stream) {
  const float* x     = (const float*)d_in[0];
  const float* pos   = (const float*)d_in[1];
  const int*   qidx  = (const int*)  d_in[2];
  const float* qattr = (const float*)d_in[3];
  const float* gamma = (const float*)d_in[4];
  const float* beta  = (const float*)d_in[5];
  const float* w1    = (const float*)d_in[6];
  const float* b1    = (const float*)d_in[7];
  const float* w2    = (const float*)d_in[8];
  const float* b2    = (const float*)d_in[9];
  const float* w3    = (const float*)d_in[10];
  const float* b3    = (const float*)d_in[11];
  const float* w_out = (const float*)d_in[12];

  float* sum_j = (float*)d_ws;                         // N_NODES*64
  float* sum_k = sum_j + (size_t)N_NODES * SCALAR_DIM; // N_NODES*64
  float* cnt_j = sum_k + (size_t)N_NODES * SCALAR_DIM; // N_NODES
  float* cnt_k = cnt_j + (size_t)N_NODES;              // N_NODES

  long zn = 2L * N_NODES * SCALAR_DIM + 2L * N_NODES;
  zero_ws_kernel<<<2048, 256, 0, stream>>>((float*)d_ws, zn);

  int dih_tiles  = (N_DIH + 15) / 16;                        // 62500
  int dih_blocks = (dih_tiles + WAVES * TPW - 1) / (WAVES * TPW); // 3907
  dih_msg_kernel<<<dih_blocks, BLOCK, 0, stream>>>(
      x, pos, qidx, qattr, gamma, beta,
      w1, b1, w2, b2, w3, b3,
      sum_j, sum_k, cnt_j, cnt_k);

  int node_tiles  = (N_NODES + 15) / 16;               // 6250
  int node_blocks = (node_tiles + WAVES - 1) / WAVES;  // 1563
  node_out_kernel<<<node_blocks, BLOCK, 0, stream>>>(
      sum_j, sum_k, cnt_j, cnt_k, w_out, (float*)d_out);
}